// HandGNNEncoder_59605556134232
// MI455X (gfx1250) — compile-verified
//
#include <hip/hip_runtime.h>

typedef __attribute__((ext_vector_type(2))) float v2f;
typedef __attribute__((ext_vector_type(8))) float v8f;

// ---- Precomputed GCN normalization constants (fixed 21-node hand skeleton) ----
// deg (with self-loop): node0=1; nodes 9,13,17 = 3; all others = 2.
// A_norm[t,s] = deg[t]^-1/2 * A[t,s] * deg[s]^-1/2
// u[t] = SELF_W[t]*x[t] + sum_i IN_W[t][i]*x[IN_SRC[t][i]]
__constant__ float SELF_W[21] = {
    1.0f, 0.5f, 0.5f, 0.5f, 0.5f,
    0.5f, 0.5f, 0.5f, 0.5f, 0.33333333f,
    0.5f, 0.5f, 0.5f, 0.33333333f,
    0.5f, 0.5f, 0.5f, 0.33333333f,
    0.5f, 0.5f, 0.5f };
__constant__ int IN_SRC[21][2] = {
    {-1,-1},{0,-1},{1,-1},{2,-1},{3,-1},
    {0,-1},{5,-1},{6,-1},{7,-1},{0,5},
    {9,-1},{10,-1},{11,-1},{0,9},
    {13,-1},{14,-1},{15,-1},{0,13},
    {17,-1},{18,-1},{19,-1} };
__constant__ float IN_W[21][2] = {
    {0.f,0.f},{0.70710678f,0.f},{0.5f,0.f},{0.5f,0.f},{0.5f,0.f},
    {0.70710678f,0.f},{0.5f,0.f},{0.5f,0.f},{0.5f,0.f},{0.57735027f,0.40824829f},
    {0.40824829f,0.f},{0.5f,0.f},{0.5f,0.f},{0.57735027f,0.33333333f},
    {0.40824829f,0.f},{0.5f,0.f},{0.5f,0.f},{0.57735027f,0.33333333f},
    {0.40824829f,0.f},{0.5f,0.f},{0.5f,0.f} };
// c[s] = (1/21) * sum_t A_norm[t,s]   (mean-pool folded through layer-2 aggregation)
__constant__ float C_POOL[21] = {
    0.19744116f, 0.04761905f, 0.04761905f, 0.04761905f, 0.02380952f,
    0.06705944f, 0.04761905f, 0.04761905f, 0.02380952f, 0.05118648f,
    0.04761905f, 0.04761905f, 0.02380952f, 0.05118648f,
    0.04761905f, 0.04761905f, 0.02380952f, 0.03531341f,
    0.04761905f, 0.04761905f, 0.02380952f };

// One block = 16 graphs. 256 threads = 8 waves (wave32).
__global__ __launch_bounds__(256) void hand_gnn_fused(
    const float* __restrict__ x,    // [G,21,2]
    const float* __restrict__ W1,   // [2,64]
    const float* __restrict__ b1,   // [64]
    const float* __restrict__ W2,   // [64,128]
    const float* __restrict__ b2,   // [128]
    float* __restrict__ out)        // [G,128]
{
    __shared__ float xs[16 * 42];
    __shared__ float us[16 * 42];
    __shared__ float vs[16][68];    // V tile [16 graphs][64 feats], row-padded

    const int tid = threadIdx.x;
    const int g0  = blockIdx.x * 16;

    // ---- Phase 1: coalesced load of 16 graphs (672 contiguous floats) ----
    const float* xg = x + (size_t)g0 * 42;
    for (int i = tid; i < 672; i += 256) xs[i] = xg[i];
    __syncthreads();

    // ---- Phase 2: u = A_norm @ x  (sparse, per (graph,node,dim)) ----
    for (int i = tid; i < 672; i += 256) {
        int g = i / 42;
        int r = i - g * 42;
        int t = r >> 1;
        int d = r & 1;
        const float* xb = &xs[g * 42];
        float u = SELF_W[t] * xb[t * 2 + d];
        int s0 = IN_SRC[t][0];
        if (s0 >= 0) u = fmaf(IN_W[t][0], xb[s0 * 2 + d], u);
        int s1 = IN_SRC[t][1];
        if (s1 >= 0) u = fmaf(IN_W[t][1], xb[s1 * 2 + d], u);
        us[i] = u;
    }
    __syncthreads();

    // ---- Phase 3: v[g][f] = sum_s c[s] * relu(u[g][s]·W1[:,f] + b1[f]) ----
    {
        int g  = tid >> 4;          // 0..15
        int fb = (tid & 15) * 4;    // 4 features per thread
        float w0[4], w1[4], bb[4], acc[4];
        #pragma unroll
        for (int j = 0; j < 4; j++) {
            w0[j] = W1[fb + j];
            w1[j] = W1[64 + fb + j];
            bb[j] = b1[fb + j];
            acc[j] = 0.0f;
        }
        const float* ub = &us[g * 42];
        #pragma unroll
        for (int s = 0; s < 21; s++) {
            float u0 = ub[s * 2 + 0];
            float u1 = ub[s * 2 + 1];
            float cs = C_POOL[s];
            #pragma unroll
            for (int j = 0; j < 4; j++) {
                float z = fmaf(u0, w0[j], fmaf(u1, w1[j], bb[j]));
                acc[j] = fmaf(cs, fmaxf(z, 0.0f), acc[j]);
            }
        }
        #pragma unroll
        for (int j = 0; j < 4; j++) vs[g][fb + j] = acc[j];
    }
    __syncthreads();

    // ---- Phase 4: out[16,128] = V[16,64] @ W2[64,128] + b2 via f32 WMMA ----
    // Wave w computes the 16-column stripe [n0, n0+16). K-loop: 16 x (16x16x4).
    {
        int wave = tid >> 5;
        int lane = tid & 31;
        int half = lane >> 4;   // 0: lanes 0-15, 1: lanes 16-31
        int l    = lane & 15;
        int n0   = wave * 16;

        v8f acc = {};
        #pragma unroll
        for (int kk = 0; kk < 16; kk++) {
            // A fragment: 16x4 f32. VGPR0 = K even, VGPR1 = K odd; hi-half lanes K+2.
            int ka = kk * 4 + half * 2;
            v2f a;
            a.x = vs[l][ka + 0];
            a.y = vs[l][ka + 1];
            // B fragment: 4x16 f32, rows striped across lanes (mirrors A's K split).
            v2f b;
            b.x = W2[(size_t)(ka + 0) * 128 + n0 + l];
            b.y = W2[(size_t)(ka + 1) * 128 + n0 + l];
            acc = __builtin_amdgcn_wmma_f32_16x16x4_f32(
                /*neg_a=*/false, a, /*neg_b=*/false, b,
                /*c_mod=*/(short)0, acc, /*reuse_a=*/false, /*reuse_b=*/false);
        }

        float bias = b2[n0 + l];
        float* ob = out + (size_t)g0 * 128;
        #pragma unroll
        for (int r = 0; r < 8; r++) {
            int m = r + half * 8;   // C/D layout: VGPR r -> M=r (lo half) / M=r+8 (hi half)
            ob[(size_t)m * 128 + n0 + l] = acc[r] + bias;
        }
    }
}

extern "C" void kernel_launch(void* const* d_in, const int* in_sizes, int n_in,
                              void* d_out, int out_size, void* d_ws, size_t ws_size,
                              hipStream_t stream) {
    const float* x  = (const float*)d_in[0];  // [B,S,21,2]
    const float* W1 = (const float*)d_in[1];  // [2,64]
    const float* b1 = (const float*)d_in[2];  // [64]
    const float* W2 = (const float*)d_in[3];  // [64,128]
    const float* b2 = (const float*)d_in[4];  // [128]
    float* out = (float*)d_out;               // [B*S,128]

    int G = in_sizes[0] / 42;                 // 21 nodes * 2 coords
    int blocks = G / 16;                      // G = 32768 -> 2048 blocks
    hand_gnn_fused<<<blocks, 256, 0, stream>>>(x, W1, b1, W2, b2, out);
}